// RGATLayer_26207890440729
// MI455X (gfx1250) — compile-verified
//
#include <hip/hip_runtime.h>
#include <hip/hip_bf16.h>

typedef __attribute__((ext_vector_type(2))) float v2f;
typedef __attribute__((ext_vector_type(4))) float v4f;
typedef __attribute__((ext_vector_type(8))) float v8f;

#define IN_DIM   64
#define OUT_DIM  64
#define NUM_RELS 8
#define RO_STRIDE (NUM_RELS * OUT_DIM)   // 512 floats per node in Z

// ---------------------------------------------------------------------------
// Kernel 0: zero h_out, denom, and the segment-max array (uint 0 == +0.0f,
// valid init since e >= 0 after ReLU, so uint ordering == float ordering).
// ---------------------------------------------------------------------------
__global__ __launch_bounds__(256) void rgat_init_kernel(unsigned* __restrict__ m,
                                                        float* __restrict__ denom,
                                                        float* __restrict__ hout,
                                                        int N) {
    int i = blockIdx.x * blockDim.x + threadIdx.x;
    if (i < N) { m[i] = 0u; denom[i] = 0.0f; }
    if (i < N * OUT_DIM) hout[i] = 0.0f;
}

// ---------------------------------------------------------------------------
// Kernel 1: Z[n][r][o] = sum_k h[n][k] * fc[r][k][o] via V_WMMA_F32_16X16X4_F32.
// Block = 256 threads (8 waves), 128 nodes per block (16 per wave).
// ALL 8 relations per block: h row fragments are loaded once into VGPRs and
// reused for 8 WMMA passes against 8 weight matrices staged in LDS (128 KB).
//
// LDS packing: Bs[r][p][n][j] = fc[r][2p+j][n]  (j in {0,1}) so each B operand
// {B[k][n], B[k+1][n]} is one aligned ds_load_b64 -> no register shuffling.
// ---------------------------------------------------------------------------
__global__ __launch_bounds__(256) void rgat_z_kernel(const float* __restrict__ h,
                                                     const float* __restrict__ fc,
                                                     float* __restrict__ Z,
                                                     int N) {
    __shared__ float Bs[NUM_RELS * IN_DIM * OUT_DIM];   // 8 * 4096 floats = 128 KB

    // Pack all 8 relation weight matrices into k-pair-interleaved LDS layout.
    // 16384 (r,p,n) pairs over 256 threads; consecutive idx -> consecutive n,
    // so the two global b32 reads per pair are row-coalesced across the wave.
    for (int idx = threadIdx.x; idx < NUM_RELS * 32 * 64; idx += 256) {
        const int r   = idx >> 11;          // 2048 pairs per relation
        const int rem = idx & 2047;
        const int p   = rem >> 6;           // k-pair index 0..31
        const int n   = rem & 63;           // output column
        const float* f = fc + ((size_t)r << 12);
        v2f b;
        b.x = f[(2 * p + 0) * OUT_DIM + n];
        b.y = f[(2 * p + 1) * OUT_DIM + n];
        *(v2f*)(&Bs[(r << 12) + (p << 7) + n * 2]) = b;
    }
    __syncthreads();

    const int wave = threadIdx.x >> 5;
    const int lane = threadIdx.x & 31;
    const int row  = lane & 15;       // M (for A) / N (for B) within 16
    const int g    = lane >> 4;       // selects K pair {0,1} vs {2,3}

    const int node0 = blockIdx.x * 128 + wave * 16;
    const int node  = node0 + row;
    const int nclamp = node < N ? node : (N - 1);   // keep EXEC all-ones for WMMA
    const float* hrow = h + (size_t)nclamp * IN_DIM;

    // Preload this lane's A fragments for all 16 k-steps (32 floats, read once).
    float a[32];
#pragma unroll
    for (int t = 0; t < 16; ++t) {
        v2f av = *(const v2f*)(hrow + t * 4 + 2 * g);
        a[2 * t + 0] = av.x;
        a[2 * t + 1] = av.y;
    }

#pragma unroll 1
    for (int r = 0; r < NUM_RELS; ++r) {
        const float* Br = Bs + (r << 12);
        v8f acc[4] = {v8f{}, v8f{}, v8f{}, v8f{}};

#pragma unroll
        for (int t = 0; t < 16; ++t) {          // K = 64 = 16 steps of 4
            const int p = t * 2 + g;            // k-pair index for this lane
            v2f av;
            av.x = a[2 * t + 0];
            av.y = a[2 * t + 1];
#pragma unroll
            for (int n0 = 0; n0 < 4; ++n0) {    // four 16-col output tiles
                v2f b = *(const v2f*)(Br + (p << 7) + (n0 * 16 + row) * 2);
                acc[n0] = __builtin_amdgcn_wmma_f32_16x16x4_f32(
                    false, av, false, b, (short)0, acc[n0], false, false);
            }
        }

        // D layout: VGPR i -> M = i + 8*g, N = row (+ n0*16)
#pragma unroll
        for (int n0 = 0; n0 < 4; ++n0) {
#pragma unroll
            for (int i = 0; i < 8; ++i) {
                int nd = node0 + g * 8 + i;
                if (nd < N)
                    Z[(size_t)nd * RO_STRIDE + r * OUT_DIM + n0 * 16 + row] = acc[n0][i];
            }
        }
    }
}

// ---------------------------------------------------------------------------
// Kernel 2: per-edge attention logit e = relu(z_s . aw[:64] + z_d . aw[64:])
// One wave per edge: lanes 0-15 cover z_s (4 floats each, b128 loads),
// lanes 16-31 cover z_d. Cross-lane reduce, then atomic segment-max on dst.
// ---------------------------------------------------------------------------
__global__ __launch_bounds__(256) void rgat_escore_kernel(const float* __restrict__ Z,
                                                          const float* __restrict__ aw,
                                                          const int* __restrict__ src,
                                                          const int* __restrict__ dst,
                                                          const int* __restrict__ etype,
                                                          float* __restrict__ e_out,
                                                          unsigned* __restrict__ m,
                                                          int E) {
    const int eid  = (blockIdx.x * blockDim.x + threadIdx.x) >> 5;
    const int lane = threadIdx.x & 31;
    if (eid >= E) return;

    const int s = src[eid], d = dst[eid], r = etype[eid];
    const int nd   = (lane < 16) ? s : d;
    const int o    = (lane & 15) * 4;
    const int awof = (lane < 16) ? 0 : OUT_DIM;

    v4f zv = *(const v4f*)(Z + (size_t)nd * RO_STRIDE + r * OUT_DIM + o);
    v4f av = *(const v4f*)(aw + r * (2 * OUT_DIM) + awof + o);
    float p = zv.x * av.x + zv.y * av.y + zv.z * av.z + zv.w * av.w;

#pragma unroll
    for (int off = 16; off > 0; off >>= 1)
        p += __shfl_xor(p, off, 32);

    if (lane == 0) {
        float ev = fmaxf(p, 0.0f);
        e_out[eid] = ev;
        atomicMax(m + d, __float_as_uint(ev));   // valid: ev >= 0
    }
}

// ---------------------------------------------------------------------------
// Kernel 3: ex = exp(e - m[dst]); denom[dst] += ex (one thread per edge).
// ---------------------------------------------------------------------------
__global__ __launch_bounds__(256) void rgat_exp_kernel(const float* __restrict__ e,
                                                       const unsigned* __restrict__ m,
                                                       const int* __restrict__ dst,
                                                       float* __restrict__ ex,
                                                       float* __restrict__ denom,
                                                       int E) {
    int i = blockIdx.x * blockDim.x + threadIdx.x;
    if (i >= E) return;
    int d = dst[i];
    float x = __expf(e[i] - __uint_as_float(m[d]));
    ex[i] = x;
    atomicAdd(denom + d, x);
}

// ---------------------------------------------------------------------------
// Kernel 4: h_out[dst] += (ex/denom[dst]) * z_s. One wave per edge, each lane
// handles 2 of 64 output dims via global_atomic_add_f32.
// ---------------------------------------------------------------------------
__global__ __launch_bounds__(256) void rgat_scatter_kernel(const float* __restrict__ Z,
                                                           const int* __restrict__ src,
                                                           const int* __restrict__ dst,
                                                           const int* __restrict__ etype,
                                                           const float* __restrict__ ex,
                                                           const float* __restrict__ denom,
                                                           float* __restrict__ hout,
                                                           int E) {
    const int eid  = (blockIdx.x * blockDim.x + threadIdx.x) >> 5;
    const int lane = threadIdx.x & 31;
    if (eid >= E) return;

    const int s = src[eid], d = dst[eid], r = etype[eid];
    const float alpha = ex[eid] / denom[d];
    const float* zs = Z + (size_t)s * RO_STRIDE + r * OUT_DIM;
    float* ho = hout + (size_t)d * OUT_DIM;

    atomicAdd(ho + lane,      alpha * zs[lane]);
    atomicAdd(ho + lane + 32, alpha * zs[lane + 32]);
}

// ---------------------------------------------------------------------------
extern "C" void kernel_launch(void* const* d_in, const int* in_sizes, int n_in,
                              void* d_out, int out_size, void* d_ws, size_t ws_size,
                              hipStream_t stream) {
    const float* h     = (const float*)d_in[0];   // (N, 64)
    const float* fc    = (const float*)d_in[1];   // (8, 64, 64)
    const float* aw    = (const float*)d_in[2];   // (8, 128, 1)
    const int*   src   = (const int*)d_in[3];     // (E,)
    const int*   dst   = (const int*)d_in[4];     // (E,)
    const int*   etype = (const int*)d_in[5];     // (E,)

    const int N = in_sizes[0] / IN_DIM;
    const int E = in_sizes[3];

    float* hout = (float*)d_out;

    // workspace layout
    float*    Z     = (float*)d_ws;               // N * 512
    float*    e     = Z + (size_t)N * RO_STRIDE;  // E
    float*    ex    = e + E;                      // E
    float*    denom = ex + E;                     // N
    unsigned* m     = (unsigned*)(denom + N);     // N

    // 0: init outputs / reduction buffers
    {
        int total = N * OUT_DIM;
        rgat_init_kernel<<<(total + 255) / 256, 256, 0, stream>>>(m, denom, hout, N);
    }
    // 1: Z = h @ fc[r] for all relations (WMMA f32 16x16x4, h read once)
    {
        dim3 grid((N + 127) / 128);
        rgat_z_kernel<<<grid, 256, 0, stream>>>(h, fc, Z, N);
    }
    // 2: edge logits + segment max
    {
        long long threads = (long long)E * 32;
        rgat_escore_kernel<<<(unsigned)((threads + 255) / 256), 256, 0, stream>>>(
            Z, aw, src, dst, etype, e, m, E);
    }
    // 3: exp + segment sum of denominators
    rgat_exp_kernel<<<(E + 255) / 256, 256, 0, stream>>>(e, m, dst, ex, denom, E);
    // 4: weighted scatter-add into h_out
    {
        long long threads = (long long)E * 32;
        rgat_scatter_kernel<<<(unsigned)((threads + 255) / 256), 256, 0, stream>>>(
            Z, src, dst, etype, ex, denom, hout, E);
    }
}